// MultiHeadAttention_75548474737193
// MI455X (gfx1250) — compile-verified
//
#include <hip/hip_runtime.h>
#include <math.h>

// Problem constants (match reference)
#define BB 16
#define TT 4
#define DD 1024
#define HH 16
#define DHD 64
#define SS 8192
#define SKK (SS + TT)          // 8196 total KV rows
#define NCH ((SKK + 15) / 16)  // 513 chunks of 16 rows
#define NSPLIT 4               // flash-decode KV splits per (b,h)

typedef __attribute__((ext_vector_type(2))) float v2f;
typedef __attribute__((ext_vector_type(8))) float v8f;

__device__ __forceinline__ v8f wmma_f32(v2f a, v2f b, v8f c) {
  // V_WMMA_F32_16X16X4_F32 : D[16x16] = A[16x4] * B[4x16] + C
  return __builtin_amdgcn_wmma_f32_16x16x4_f32(
      /*neg_a=*/false, a, /*neg_b=*/false, b,
      /*c_mod=*/(short)0, c, /*reuse_a=*/false, /*reuse_b=*/false);
}

// ---------------------------------------------------------------------------
// GEMM: C[64][N] = A[64][1024] @ W[N][1024]^T   (one 16x16 tile per wave)
// Used for QKV projection (N=3072) and output projection (N=1024).
// ---------------------------------------------------------------------------
__global__ __launch_bounds__(128)
void gemm_xwt_kernel(const float* __restrict__ A, const float* __restrict__ W,
                     float* __restrict__ C, int N) {
  const int lane = threadIdx.x & 31;
  const int wave = threadIdx.x >> 5;
  const int n    = lane & 15;   // M-index for A loads, N-index for B loads & D tile
  const int hl   = lane >> 4;   // half-wave selector (K offset 2*hl)
  const int r0   = blockIdx.y * 16;
  const int e0   = blockIdx.x * 64 + wave * 16;

  const float* arow = A + (size_t)(r0 + n) * DD + 2 * hl;
  const float* wrow = W + (size_t)(e0 + n) * DD + 2 * hl;

  v8f acc = {};
#pragma unroll 8
  for (int ks = 0; ks < DD / 4; ++ks) {
    v2f a = *(const v2f*)(arow + 4 * ks);   // A[m][4ks + j + 2*hl]
    v2f b = *(const v2f*)(wrow + 4 * ks);   // B[4ks + j + 2*hl][n] = W[e0+n][...]
    acc = wmma_f32(a, b, acc);
  }
#pragma unroll
  for (int r = 0; r < 8; ++r)
    C[(size_t)(r0 + r + 8 * hl) * N + e0 + n] = acc[r];
}

// ---------------------------------------------------------------------------
// Flash-decode attention, split-KV: block = (b,h) x split; 8 waves per block
// stream a strided subset of the 16-row KV chunks. Each block writes
// UNNORMALIZED partials (O, m, l) for its split to workspace.
// qkv layout: [B*T][3072] rows = {q | k_new | v_new}.
// ---------------------------------------------------------------------------
__global__ __launch_bounds__(256)
void attn_split_kernel(const float* __restrict__ cache_k,
                       const float* __restrict__ cache_v,
                       const float* __restrict__ qkv,
                       float* __restrict__ part_o,   // [B*H][NSPLIT][T][64]
                       float* __restrict__ part_m,   // [B*H][NSPLIT][T]
                       float* __restrict__ part_l) { // [B*H][NSPLIT][T]
  __shared__ float ldsV[8][16 * 80];   // per-wave V chunk, stride 80 (==16 mod 64 banks)
  __shared__ float red_o[8][4][64];
  __shared__ float red_m[8][4];
  __shared__ float red_l[8][4];

  const int bh    = blockIdx.x;
  const int split = blockIdx.y;
  const int b     = bh / HH;
  const int h     = bh % HH;
  const int wave  = threadIdx.x >> 5;
  const int lane  = threadIdx.x & 31;
  const int n     = lane & 15;
  const int hl    = lane >> 4;

  // Preload Q^T operand tiles (loop-invariant), pre-scaled by 1/sqrt(64).
  v2f qB[16];
  {
    const int qn = (n < TT) ? n : 0;
    const float qs = (n < TT) ? 0.125f : 0.0f;
    const float* qrow = qkv + (size_t)(b * TT + qn) * 3072 + h * DHD + 2 * hl;
#pragma unroll
    for (int ks = 0; ks < 16; ++ks) {
      qB[ks].x = qrow[4 * ks + 0] * qs;
      qB[ks].y = qrow[4 * ks + 1] * qs;
    }
  }

  float m = -1e30f, l = 0.0f;
  v8f oacc[4] = {};   // out^T tiles: lane holds (dim = 16t + r + 8*hl, query = n)

  for (int c = split * 8 + wave; c < NCH; c += 8 * NSPLIT) {
    const int rb = c * 16;

    // Per-lane K row pointer (lane n -> chunk-local row n). Rows >= S come
    // from the freshly projected k_new inside qkv; OOB rows clamped+masked.
    const float* kp;
    {
      int row = rb + n;
      int rc = row < SKK ? row : SKK - 1;
      if (rc < SS) kp = cache_k + ((size_t)(b * SS + rc) * HH + h) * DHD;
      else         kp = qkv + (size_t)(b * TT + (rc - SS)) * 3072 + 1024 + h * DHD;
      kp += 2 * hl;
    }

    // Stage V chunk into LDS (coalesced 16B per lane, 2 rows per iteration).
#pragma unroll
    for (int i = 0; i < 8; ++i) {
      int r = 2 * i + hl;
      int row = rb + r;
      int rc = row < SKK ? row : SKK - 1;
      const float* vp;
      if (rc < SS) vp = cache_v + ((size_t)(b * SS + rc) * HH + h) * DHD;
      else         vp = qkv + (size_t)(b * TT + (rc - SS)) * 3072 + 2048 + h * DHD;
      const float4 vv = ((const float4*)vp)[n];
      *(float4*)&ldsV[wave][r * 80 + 4 * n] = vv;
    }

    // Scores^T: S[16 rows][16 queries] = K_chunk[16x64] @ Q^T[64x16]
    v8f s = {};
#pragma unroll
    for (int ks = 0; ks < 16; ++ks) {
      v2f a = *(const v2f*)(kp + 4 * ks);
      s = wmma_f32(a, qB[ks], s);
    }

    // Online softmax (per query column n; halves combined via xor-16 shuffle).
    float pv[8];
    float cm = -1e30f;
#pragma unroll
    for (int r = 0; r < 8; ++r) {
      int row = rb + r + 8 * hl;
      float sv = (row < SKK) ? s[r] : -1e30f;
      pv[r] = sv;
      cm = fmaxf(cm, sv);
    }
    cm = fmaxf(cm, __shfl_xor(cm, 16, 32));
    const float nm = fmaxf(m, cm);
    const float alpha = __expf(m - nm);
    float rsum = 0.0f;
#pragma unroll
    for (int r = 0; r < 8; ++r) { pv[r] = __expf(pv[r] - nm); rsum += pv[r]; }
    rsum += __shfl_xor(rsum, 16, 32);
    l = l * alpha + rsum;
    m = nm;
#pragma unroll
    for (int t = 0; t < 4; ++t) oacc[t] *= alpha;

    // Build P operand tiles: B[4k+j+2hl][n] = p[row][n]; halves swapped via xor-16.
    float ov[8];
#pragma unroll
    for (int r = 0; r < 8; ++r) ov[r] = __shfl_xor(pv[r], 16, 32);
    v2f pB[4];
#pragma unroll
    for (int k2 = 0; k2 < 4; ++k2) {
      int r0i = 4 * k2 + 2 * hl;
      int r1i = r0i + 1;
      pB[k2].x = ((r0i >> 3) == hl) ? pv[r0i & 7] : ov[r0i & 7];
      pB[k2].y = ((r1i >> 3) == hl) ? pv[r1i & 7] : ov[r1i & 7];
    }

    // out^T[64x16] += V^T[64x16rows] @ P[16rows x 16]; A from LDS (transposed V).
#pragma unroll
    for (int t = 0; t < 4; ++t) {
#pragma unroll
      for (int k2 = 0; k2 < 4; ++k2) {
        v2f a;
        a.x = ldsV[wave][(4 * k2 + 2 * hl + 0) * 80 + 16 * t + n];
        a.y = ldsV[wave][(4 * k2 + 2 * hl + 1) * 80 + 16 * t + n];
        oacc[t] = wmma_f32(a, pB[k2], oacc[t]);
      }
    }
  }

  // Publish per-wave partials.
  if (n < TT) {
#pragma unroll
    for (int t = 0; t < 4; ++t)
#pragma unroll
      for (int r = 0; r < 8; ++r)
        red_o[wave][n][16 * t + r + 8 * hl] = oacc[t][r];
    if (hl == 0) { red_m[wave][n] = m; red_l[wave][n] = l; }
  }
  __syncthreads();

  // Merge the 8 wave partials of this split (still unnormalized) -> workspace.
  {
    const int q = threadIdx.x >> 6;   // 0..3
    const int d = threadIdx.x & 63;   // 0..63
    float Mb = -1e30f;
#pragma unroll
    for (int w = 0; w < 8; ++w) Mb = fmaxf(Mb, red_m[w][q]);
    float Lb = 0.0f, Ob = 0.0f;
#pragma unroll
    for (int w = 0; w < 8; ++w) {
      float sc = __expf(red_m[w][q] - Mb);
      Lb += red_l[w][q] * sc;
      Ob += red_o[w][q][d] * sc;
    }
    const int pidx = bh * NSPLIT + split;
    part_o[((size_t)pidx * TT + q) * DHD + d] = Ob;
    if (d == 0) {
      part_m[pidx * TT + q] = Mb;
      part_l[pidx * TT + q] = Lb;
    }
  }
}

// ---------------------------------------------------------------------------
// Combine the NSPLIT split partials per (b,h) with a log-sum-exp merge.
// ---------------------------------------------------------------------------
__global__ __launch_bounds__(256)
void attn_combine_kernel(const float* __restrict__ part_o,
                         const float* __restrict__ part_m,
                         const float* __restrict__ part_l,
                         float* __restrict__ attn_out) {
  const int bh = blockIdx.x;
  const int b  = bh / HH;
  const int h  = bh % HH;
  const int q  = threadIdx.x >> 6;
  const int d  = threadIdx.x & 63;

  float M = -1e30f;
#pragma unroll
  for (int s = 0; s < NSPLIT; ++s)
    M = fmaxf(M, part_m[(bh * NSPLIT + s) * TT + q]);
  float num = 0.0f, den = 0.0f;
#pragma unroll
  for (int s = 0; s < NSPLIT; ++s) {
    const int pidx = bh * NSPLIT + s;
    const float sc = __expf(part_m[pidx * TT + q] - M);
    num += part_o[((size_t)pidx * TT + q) * DHD + d] * sc;
    den += part_l[pidx * TT + q] * sc;
  }
  attn_out[(size_t)(b * TT + q) * DD + h * DHD + d] = num / den;
}

// ---------------------------------------------------------------------------
extern "C" void kernel_launch(void* const* d_in, const int* in_sizes, int n_in,
                              void* d_out, int out_size, void* d_ws, size_t ws_size,
                              hipStream_t stream) {
  const float* x       = (const float*)d_in[0];  // [16,4,1024]
  const float* cache_k = (const float*)d_in[1];  // [16,8192,16,64]
  const float* cache_v = (const float*)d_in[2];
  const float* w_qkv   = (const float*)d_in[3];  // [3072,1024]
  const float* w_proj  = (const float*)d_in[4];  // [1024,1024]
  float* out = (float*)d_out;                    // [16,4,1024]

  float* ws       = (float*)d_ws;
  float* qkv      = ws;                                   // [64][3072]
  float* attn_out = qkv + (size_t)64 * 3072;              // [64][1024]
  float* part_o   = attn_out + (size_t)64 * 1024;         // [256][NSPLIT][4][64]
  float* part_m   = part_o + (size_t)BB * HH * NSPLIT * TT * DHD;  // [256][NSPLIT][4]
  float* part_l   = part_m + (size_t)BB * HH * NSPLIT * TT;

  // 1) QKV projection: qkv[64][3072] = x[64][1024] @ w_qkv^T
  gemm_xwt_kernel<<<dim3(3072 / 64, 4), 128, 0, stream>>>(x, w_qkv, qkv, 3072);

  // 2) Flash-decode attention, split-KV for occupancy/MLP (1024 blocks)
  attn_split_kernel<<<dim3(BB * HH, NSPLIT), 256, 0, stream>>>(
      cache_k, cache_v, qkv, part_o, part_m, part_l);

  // 2b) Log-sum-exp combine of the splits
  attn_combine_kernel<<<dim3(BB * HH), 256, 0, stream>>>(
      part_o, part_m, part_l, attn_out);

  // 3) Output projection: out[64][1024] = attn_out @ w_proj^T
  gemm_xwt_kernel<<<dim3(1024 / 64, 4), 128, 0, stream>>>(attn_out, w_proj, out, 1024);
}